// BbProjection_27358941676275
// MI455X (gfx1250) — compile-verified
//
#include <hip/hip_runtime.h>

// BbProjection: out[:, :3] = clip(y_pred[:, :3], lx, ux)
//               out[:, 3:] = clip(y_pred[:, 3:], ly, uy)
// Pure streaming kernel: 64 B of HBM traffic per row, ~12 flops per row.
// Bandwidth-bound at 23.3 TB/s -> target ~11 us. No matrix math exists in
// this op, so WMMA is intentionally absent; the CDNA5 wins are b128 NT
// vector memory ops and v_med3 clamps under wave32.

typedef float v4f __attribute__((ext_vector_type(4)));

__device__ __forceinline__ float clampf(float x, float lo, float hi) {
    // med3(x, lo, hi) == clip(x, lo, hi) when lo <= hi (guaranteed by setup:
    // ux = lx + positive, uy = ly + positive). Single VALU op on gfx1250.
    return __builtin_amdgcn_fmed3f(x, lo, hi);
}

__global__ __launch_bounds__(256) void bb_projection_kernel(
    const v4f* __restrict__ yp,   // y_pred viewed as float4; 3 per row-pair
    const v4f* __restrict__ cp,   // constr_para viewed as float4; 1 per row
    v4f* __restrict__ out,        // output viewed as float4; 3 per row-pair
    int npairs)                   // number of row pairs (rows/2)
{
    int t = blockIdx.x * blockDim.x + threadIdx.x;
    if (t >= npairs) return;

    // Two rows per thread: 12 y_pred floats (48 B, 16-B aligned since
    // 48*t % 16 == 0) and 8 bound floats (32 B, aligned).
    long base3 = 3L * t;
    long base2 = 2L * t;

    v4f a  = __builtin_nontemporal_load(yp + base3 + 0); // r0: c0 c1 c2 c3
    v4f b  = __builtin_nontemporal_load(yp + base3 + 1); // r0: c4 c5 | r1: c0 c1
    v4f c  = __builtin_nontemporal_load(yp + base3 + 2); // r1: c2 c3 c4 c5
    v4f p0 = __builtin_nontemporal_load(cp + base2 + 0); // r0: lx ux ly uy
    v4f p1 = __builtin_nontemporal_load(cp + base2 + 1); // r1: lx ux ly uy

    v4f ra, rb, rc;
    // row 0: cols 0..2 -> [lx0,ux0], cols 3..5 -> [ly0,uy0]
    ra.x = clampf(a.x, p0.x, p0.y);
    ra.y = clampf(a.y, p0.x, p0.y);
    ra.z = clampf(a.z, p0.x, p0.y);
    ra.w = clampf(a.w, p0.z, p0.w);
    rb.x = clampf(b.x, p0.z, p0.w);
    rb.y = clampf(b.y, p0.z, p0.w);
    // row 1: cols 0..2 -> [lx1,ux1], cols 3..5 -> [ly1,uy1]
    rb.z = clampf(b.z, p1.x, p1.y);
    rb.w = clampf(b.w, p1.x, p1.y);
    rc.x = clampf(c.x, p1.x, p1.y);
    rc.y = clampf(c.y, p1.z, p1.w);
    rc.z = clampf(c.z, p1.z, p1.w);
    rc.w = clampf(c.w, p1.z, p1.w);

    __builtin_nontemporal_store(ra, out + base3 + 0);
    __builtin_nontemporal_store(rb, out + base3 + 1);
    __builtin_nontemporal_store(rc, out + base3 + 2);
}

extern "C" void kernel_launch(void* const* d_in, const int* in_sizes, int n_in,
                              void* d_out, int out_size, void* d_ws, size_t ws_size,
                              hipStream_t stream) {
    const v4f* yp = (const v4f*)d_in[0];   // y_pred, in_sizes[0] = rows*6 floats
    const v4f* cp = (const v4f*)d_in[1];   // constr_para, rows*4 floats
    v4f* out = (v4f*)d_out;                // rows*6 floats

    int rows   = in_sizes[0] / 6;          // 4,000,000 (even)
    int npairs = rows / 2;                 // 2,000,000 row-pairs

    const int threads = 256;               // 8 wave32 waves per block
    int blocks = (npairs + threads - 1) / threads;

    bb_projection_kernel<<<blocks, threads, 0, stream>>>(yp, cp, out, npairs);
}